// ConvDGN_9612136808453
// MI455X (gfx1250) — compile-verified
//
#include <hip/hip_runtime.h>
#include <stdint.h>

// ---------------------------------------------------------------------------
// 2-layer GCN for MI455X (gfx1250, wave32).
// GEMMs use V_WMMA_F32_16X16X4_F32 (fp32-exact, matches fp32 reference).
// Edge aggregation: wave-per-edge b128 gathers + native global_atomic_add_f32
// (inline asm, device scope) -- scatter targets are L2-resident (192 MB L2).
// ---------------------------------------------------------------------------

typedef __attribute__((ext_vector_type(2))) float v2f;
typedef __attribute__((ext_vector_type(8))) float v8f;

// Guaranteed hardware f32 atomic (no CAS-loop fallback). Non-returning form,
// device scope so cross-workgroup scatter-adds are coherent at L2.
__device__ __forceinline__ void atomic_add_f32(float* p, float v) {
  asm volatile("global_atomic_add_f32 %0, %1, off scope:SCOPE_DEV"
               :: "v"((uint64_t)(uintptr_t)p), "v"(v)
               : "memory");
}

// ---------------- degree / normalization ----------------

__global__ void k_deg_init(float* __restrict__ deg, int n) {
  int i = blockIdx.x * blockDim.x + threadIdx.x;
  if (i < n) deg[i] = 1.0f;  // self-loop weight
}

__global__ void k_deg_accum(const long long* __restrict__ col,
                            const float* __restrict__ ew,
                            float* __restrict__ deg, long long E) {
  long long i = (long long)blockIdx.x * blockDim.x + threadIdx.x;
  if (i < E) atomic_add_f32(&deg[col[i]], ew[i]);
}

__global__ void k_deg_to_dis(float* __restrict__ deg, int n) {
  int i = blockIdx.x * blockDim.x + threadIdx.x;
  if (i < n) {
    float d = deg[i];
    deg[i] = (d > 0.0f) ? rsqrtf(d) : 0.0f;  // deg >= 1 here, guard for parity
  }
}

// ---------------- WMMA f32 GEMM: C[M,NCOL] = A[M,K] * B[K,NCOL] ----------------
// One wave computes one 16x16 output tile; 8 waves per 256-thread block cover
// 128 consecutive rows. K is stepped 4 at a time with V_WMMA_F32_16X16X4_F32.
//
// ISA VGPR layouts (CDNA5 ISA):
//   A 16x4:  lanes 0-15 hold row m, {v0,v1}={K0,K1}; lanes 16-31 hold {K2,K3}
//   B 4x16:  v0 = B[K0 or K2][n=lane&15], v1 = next K row
//   C/D:     v[i] = row (i + 8*(lane>>4)), col = lane&15

template <int K, int NCOL>
__global__ __launch_bounds__(256) void k_gemm_wmma_f32(
    const float* __restrict__ A, const float* __restrict__ B,
    float* __restrict__ C, int M) {
  const int lane = threadIdx.x & 31;
  const int wave = threadIdx.x >> 5;
  const int half = lane >> 4;   // 0 = lanes 0-15, 1 = lanes 16-31
  const int m    = lane & 15;

  const int row0 = (blockIdx.x * 8 + wave) * 16;
  if (row0 >= M) return;                     // whole-wave uniform exit
  const int col0 = blockIdx.y * 16;

  const float* aRow = A + (size_t)(row0 + m) * K;
  v8f acc = {};

#pragma unroll
  for (int k0 = 0; k0 < K; k0 += 4) {
    // A tile: contiguous float2 per lane (8-byte aligned -> b64 load)
    v2f a = *(const v2f*)(aRow + k0 + 2 * half);
    // B tile: two rows of W, strided by NCOL
    const float* bp = B + (size_t)(k0 + 2 * half) * NCOL + col0 + m;
    v2f b;
    b.x = bp[0];
    b.y = bp[NCOL];
    acc = __builtin_amdgcn_wmma_f32_16x16x4_f32(
        /*neg_a=*/false, a, /*neg_b=*/false, b,
        /*c_mod=*/(short)0, acc, /*reuse_a=*/false, /*reuse_b=*/false);
  }

#pragma unroll
  for (int i = 0; i < 8; ++i)
    C[(size_t)(row0 + i + 8 * half) * NCOL + col0 + m] = acc[i];
}

// ---------------- self-loop seed: agg = dis^2 * h ----------------

template <int F>
__global__ void k_selfloop_init(const float* __restrict__ h,
                                const float* __restrict__ dis,
                                float* __restrict__ agg, int n) {
  constexpr int C4 = F / 4;                        // float4 chunks per row
  long long idx = (long long)blockIdx.x * blockDim.x + threadIdx.x;
  if (idx >= (long long)n * C4) return;
  int node = (int)(idx / C4);
  float s = dis[node];
  s *= s;
  float4 v = ((const float4*)h)[idx];
  v.x *= s; v.y *= s; v.z *= s; v.w *= s;
  ((float4*)agg)[idx] = v;
}

// ---------------- edge scatter: agg[c] += dis[r]*w*dis[c] * h[r] ----------------
// One wave per edge; each lane handles F/32 contiguous floats (float4/float2
// gather from the source row, per-float f32 hardware atomics to the dst row).

template <int F>
__global__ __launch_bounds__(256) void k_edge_aggregate(
    const long long* __restrict__ row, const long long* __restrict__ col,
    const float* __restrict__ ew, const float* __restrict__ dis,
    const float* __restrict__ h, float* __restrict__ agg, long long E) {
  constexpr int PER = F / 32;
  long long e = ((long long)blockIdx.x * blockDim.x + threadIdx.x) >> 5;
  if (e >= E) return;
  const int lane = threadIdx.x & 31;

  long long r = row[e];
  long long c = col[e];
  float norm = dis[r] * ew[e] * dis[c];

  const float* hs = h + (size_t)r * F + lane * PER;
  float* as = agg + (size_t)c * F + lane * PER;

  float v[PER];
#pragma unroll
  for (int j = 0; j < PER; ++j) v[j] = hs[j];     // contiguous -> b64/b128 load
#pragma unroll
  for (int j = 0; j < PER; ++j) atomic_add_f32(&as[j], norm * v[j]);
}

// ---------------- bias + ReLU (in place) ----------------

template <int F>
__global__ void k_bias_relu(float* __restrict__ agg, const float* __restrict__ b,
                            int n) {
  long long idx = (long long)blockIdx.x * blockDim.x + threadIdx.x;
  if (idx >= (long long)n * F) return;
  int f = (int)(idx % F);
  float v = agg[idx] + b[f];
  agg[idx] = v > 0.0f ? v : 0.0f;
}

// ---------------------------------------------------------------------------

static inline size_t align256(size_t x) { return (x + 255) & ~(size_t)255; }

extern "C" void kernel_launch(void* const* d_in, const int* in_sizes, int n_in,
                              void* d_out, int out_size, void* d_ws, size_t ws_size,
                              hipStream_t stream) {
  constexpr int IN_DIM = 128, HID = 128, OUT_DIM = 64;

  const float*     x   = (const float*)d_in[0];
  const long long* ei  = (const long long*)d_in[1];   // int64 [2, E]
  const float*     ew  = (const float*)d_in[2];
  const float*     W1  = (const float*)d_in[3];
  const float*     b1  = (const float*)d_in[4];
  const float*     W2  = (const float*)d_in[5];
  const float*     b2  = (const float*)d_in[6];
  float*           out = (float*)d_out;

  const int       N = in_sizes[0] / IN_DIM;
  const long long E = in_sizes[2];
  const long long* row = ei;
  const long long* col = ei + E;

  // workspace layout
  char* ws = (char*)d_ws;
  float* dis  = (float*)ws;                                  // N        (deg -> dis)
  size_t off  = align256((size_t)N * sizeof(float));
  float* h1   = (float*)(ws + off);                          // N*HID
  off += align256((size_t)N * HID * sizeof(float));
  float* agg1 = (float*)(ws + off);                          // N*HID (relu'd in place)
  off += align256((size_t)N * HID * sizeof(float));
  float* h2   = (float*)(ws + off);                          // N*OUT_DIM
  (void)ws_size;

  const int T = 256;
  auto cdiv = [](long long a, long long b) { return (int)((a + b - 1) / b); };

  // ---- normalization coefficients (shared by both layers) ----
  k_deg_init<<<cdiv(N, T), T, 0, stream>>>(dis, N);
  k_deg_accum<<<cdiv(E, T), T, 0, stream>>>(col, ew, dis, E);
  k_deg_to_dis<<<cdiv(N, T), T, 0, stream>>>(dis, N);

  const int rowTiles = cdiv(N, 16);
  const int rowBlocks = cdiv(rowTiles, 8);

  // ---- layer 1: h1 = x @ W1 ; agg1 = norm-aggregate(h1) ; relu(agg1 + b1) ----
  k_gemm_wmma_f32<IN_DIM, HID>
      <<<dim3(rowBlocks, HID / 16), T, 0, stream>>>(x, W1, h1, N);
  k_selfloop_init<HID><<<cdiv((long long)N * (HID / 4), T), T, 0, stream>>>(h1, dis, agg1, N);
  k_edge_aggregate<HID><<<cdiv(E * 32, T), T, 0, stream>>>(row, col, ew, dis, h1, agg1, E);
  k_bias_relu<HID><<<cdiv((long long)N * HID, T), T, 0, stream>>>(agg1, b1, N);

  // ---- layer 2: h2 = agg1 @ W2 ; out = norm-aggregate(h2) ; relu(out + b2) ----
  k_gemm_wmma_f32<HID, OUT_DIM>
      <<<dim3(rowBlocks, OUT_DIM / 16), T, 0, stream>>>(agg1, W2, h2, N);
  k_selfloop_init<OUT_DIM><<<cdiv((long long)N * (OUT_DIM / 4), T), T, 0, stream>>>(h2, dis, out, N);
  k_edge_aggregate<OUT_DIM><<<cdiv(E * 32, T), T, 0, stream>>>(row, col, ew, dis, h2, out, E);
  k_bias_relu<OUT_DIM><<<cdiv((long long)N * OUT_DIM, T), T, 0, stream>>>(out, b2, N);
}